// TileWarping_28681791602997
// MI455X (gfx1250) — compile-verified
//
#include <hip/hip_runtime.h>
#include <cstdint>

#define NB   8
#define NC   16
#define HH   512
#define WW   1024
#define TPC  34
#define TTH  128          // tile grid rows   (HH/4)
#define TTW  256          // tile grid cols   (WW/4)
#define RW   32           // region width per block (x)
#define RH   4            // region height per block (y) == one tile row
#define NOFF 9

typedef uint32_t u32;
typedef uint64_t u64;
typedef __attribute__((ext_vector_type(4))) u32 u32x4;
typedef __attribute__((ext_vector_type(8))) int i32x8;
typedef __attribute__((ext_vector_type(4))) int i32x4;
// 4-byte-aligned float4: lowers to global_load_b128 (CDNA5 DWORD-mode allows
// dword-aligned B128), one load per tap row instead of four.
typedef __attribute__((ext_vector_type(4), aligned(4))) float vf4;

__device__ __forceinline__ int clampi(int v, int lo, int hi) {
  return v < lo ? lo : (v > hi ? hi : v);
}

// One wave per block. Lane l -> column X0+l; each lane owns 4 rows Y0..Y0+3
// (one vertical run of a 4x4 tile, so i = p and the plane coefficients are
// shared across the lane's 4 pixels).
__global__ __launch_bounds__(32) void tile_warp_cv_kernel(
    const float* __restrict__ tp,   // (8,34,128,256)
    const float* __restrict__ fl,   // (8,16,512,1024)
    const float* __restrict__ fr,   // (8,16,512,1024)
    float* __restrict__ out)        // (8,144,128,256)
{
  __shared__ float lds_l[NC * RH * RW];   // 8 KB fea_l tile, packed [c][y][x]

  const int l  = (int)threadIdx.x;
  const int X0 = (int)blockIdx.x * RW;
  const int Y0 = (int)blockIdx.y * RH;
  const int b  = (int)blockIdx.z;

#if defined(__HIP_DEVICE_COMPILE__)
  // ---- CDNA5 TDM: async DMA fea_l[b, 0:16, Y0:Y0+4, X0:X0+32] -> LDS ----
  {
    const u64 gaddr = (u64)(uintptr_t)(fl + ((u64)b * NC * HH + (u64)Y0) * WW + X0);
    const u32 ldsa  = (u32)(uintptr_t)(&lds_l[0]);

    u32x4 g0;                                  // D# group 0 (128b)
    g0[0] = 1u;                                // count=1 (valid), no gather/restore
    g0[1] = ldsa;                              // lds_addr (bytes)
    g0[2] = (u32)(gaddr & 0xFFFFFFFFull);      // global_addr[31:0]
    g0[3] = ((u32)(gaddr >> 32) & 0x01FFFFFFu) // global_addr[56:32]
          | (2u << 30);                        // type = 2 ("image")

    const u64 s1 = (u64)HH * (u64)WW;          // dim1 (channel) stride, elements
    i32x8 g1;                                  // D# group 1 (256b)
    g1[0] = (int)(2u << 16);                                   // data_size=2 -> 4B
    g1[1] = (int)(((u32)WW & 0xFFFFu) << 16);                  // tensor_dim0 lo16
    g1[2] = (int)((((u32)WW >> 16) & 0xFFFFu)
          | (((u32)HH & 0xFFFFu) << 16));                      // dim0 hi | dim1 lo
    g1[3] = (int)((((u32)HH >> 16) & 0xFFFFu)
          | ((u32)RW << 16));                                  // dim1 hi | tile_dim0
    g1[4] = (int)((u32)RH | ((u32)NC << 16));                  // tile_dim1 | tile_dim2
    g1[5] = (int)(u32)WW;                                      // dim0_stride [31:0]
    g1[6] = (int)(((u32)(s1 & 0xFFFFull)) << 16);              // dim0_str hi | dim1_str lo16
    g1[7] = (int)(u32)(s1 >> 16);                              // dim1_stride [47:16]

    i32x4 g2; g2[0] = NC; g2[1] = 0; g2[2] = 0; g2[3] = 0;     // tensor_dim2 = 16
    i32x4 g3; g3[0] = 0;  g3[1] = 0; g3[2] = 0; g3[3] = 0;
#if defined(__clang_major__) && (__clang_major__ >= 23)
    i32x8 g4; g4[0]=0; g4[1]=0; g4[2]=0; g4[3]=0; g4[4]=0; g4[5]=0; g4[6]=0; g4[7]=0;
    __builtin_amdgcn_tensor_load_to_lds(g0, g1, g2, g3, g4, 0);
#else
    __builtin_amdgcn_tensor_load_to_lds(g0, g1, g2, g3, 0);
#endif
  }
#endif

  const int X  = X0 + l;
  const int jj = X & 3;          // sub-tile x (output channel j)
  const int tx = X >> 2;
  const int ty = Y0 >> 2;

  // Affine plane coefficients (shared by this lane's 4 pixels).
  const u64 PL = (u64)(TTH * TTW);
  const u64 tpb = (u64)b * TPC * PL + (u64)ty * TTW + (u64)tx;
  const float fxv = tp[tpb +  0 * PL];
  const float fxx = tp[tpb +  1 * PL];
  const float fxy = tp[tpb +  2 * PL];
  const float fyv = tp[tpb + 16 * PL];
  const float fyx = tp[tpb + 17 * PL];
  const float fyy = tp[tpb + 18 * PL];

  const float offj = (float)jj - 1.5f;

  // Integer flow offsets shift x0/y0 only; fractional weights are identical
  // for all 9 offsets, so one 4x4 tap window serves all of them.
  int x0[4], y0[4], off0[4];
  bool inb[4];
  float w00[4], w01[4], w10[4], w11[4];
#pragma unroll
  for (int p = 0; p < 4; ++p) {
    const float offi = (float)p - 1.5f;
    const float gx = (float)X        + fxv + offi * fxy + offj * fxx;
    const float gy = (float)(Y0 + p) + fyv + offi * fyy + offj * fyx;
    const float xf = floorf(gx);
    const float yf = floorf(gy);
    const float wx = gx - xf;
    const float wy = gy - yf;
    x0[p] = (int)xf;
    y0[p] = (int)yf;
    // All 16 taps in-bounds (clamps = identity, valid = 1)?
    inb[p]  = (x0[p] >= 1) && (x0[p] <= WW - 3) &&
              (y0[p] >= 1) && (y0[p] <= HH - 3);
    off0[p] = (y0[p] - 1) * WW + (x0[p] - 1);  // channel-invariant base offset
    const float omx = 1.0f - wx, omy = 1.0f - wy;
    w00[p] = omx * omy; w01[p] = wx * omy; w10[p] = omx * wy; w11[p] = wx * wy;
  }

  float acc[NOFF][4];
#pragma unroll
  for (int o = 0; o < NOFF; ++o)
#pragma unroll
    for (int p = 0; p < 4; ++p) acc[o][p] = 0.0f;

#if defined(__HIP_DEVICE_COMPILE__)
  __builtin_amdgcn_s_wait_tensorcnt(0);   // TDM done -> LDS tile valid
  asm volatile("" ::: "memory");
#endif

  const float* __restrict__ frb = fr + (u64)b * NC * HH * WW;

#pragma unroll 1
  for (int c = 0; c < NC; ++c) {
    const float* __restrict__ base = frb + (u64)c * (HH * WW);

    float L[4];
#pragma unroll
    for (int p = 0; p < 4; ++p) L[p] = lds_l[(c * RH + p) * RW + l];

#pragma unroll
    for (int p = 0; p < 4; ++p) {
      // 4x4 tap neighborhood shared by all 9 offsets.
      float t[4][4];
      if (inb[p]) {
        // Fast path: 4 dword-aligned b128 row loads, no clamps, no masking.
        const float* __restrict__ q = base + off0[p];
#pragma unroll
        for (int dy = 0; dy < 4; ++dy) {
          const vf4 r = *reinterpret_cast<const vf4*>(q + dy * WW);
          t[dy][0] = r.x; t[dy][1] = r.y; t[dy][2] = r.z; t[dy][3] = r.w;
        }
      } else {
        // Border path: exact reference semantics (clamp index, zero weight).
#pragma unroll
        for (int dy = 0; dy < 4; ++dy) {
          const int yi = y0[p] - 1 + dy;
          const bool vy = (yi >= 0) && (yi < HH);
          const float* __restrict__ row = base + (u64)clampi(yi, 0, HH - 1) * WW;
#pragma unroll
          for (int dx = 0; dx < 4; ++dx) {
            const int xi = x0[p] - 1 + dx;
            const bool v = vy && (xi >= 0) && (xi < WW);
            const float g = row[clampi(xi, 0, WW - 1)];
            t[dy][dx] = v ? g : 0.0f;
          }
        }
      }
#pragma unroll
      for (int ox = -1; ox <= 1; ++ox) {
#pragma unroll
        for (int oy = -1; oy <= 1; ++oy) {
          const int o = (ox + 1) * 3 + (oy + 1);   // reference append order
          const float wv = t[oy + 1][ox + 1] * w00[p]
                         + t[oy + 1][ox + 2] * w01[p]
                         + t[oy + 2][ox + 1] * w10[p]
                         + t[oy + 2][ox + 2] * w11[p];
          acc[o][p] += fabsf(L[p] - wv);
        }
      }
    }
    if (c + 1 < NC) {
      // Warm next channel's neighborhood (global_prefetch_b8): same (y,x),
      // one channel-plane stride ahead.
      __builtin_prefetch(base + (u64)(HH * WW)
                              + (u64)clampi(y0[0] - 1, 0, HH - 1) * WW
                              + clampi(x0[0] - 1, 0, WW - 1), 0, 1);
    }
  }

  // out[b, o*16 + p*4 + jj, ty, tx]
  float* __restrict__ ob = out + (u64)b * (NOFF * NC) * PL + (u64)ty * TTW + (u64)tx;
#pragma unroll
  for (int o = 0; o < NOFF; ++o)
#pragma unroll
    for (int p = 0; p < 4; ++p) {
      const int ch = o * 16 + p * 4 + jj;
      ob[(u64)ch * PL] = acc[o][p];
    }
}

extern "C" void kernel_launch(void* const* d_in, const int* in_sizes, int n_in,
                              void* d_out, int out_size, void* d_ws, size_t ws_size,
                              hipStream_t stream) {
  const float* tp = (const float*)d_in[0];   // tile_plane
  const float* fl = (const float*)d_in[1];   // fea_l
  const float* fr = (const float*)d_in[2];   // fea_r
  float* out = (float*)d_out;

  dim3 grid(WW / RW, HH / RH, NB);           // (32, 128, 8) single-wave blocks
  tile_warp_cv_kernel<<<grid, dim3(32, 1, 1), 0, stream>>>(tp, fl, fr, out);
}